// TopKRouter_10402410791601
// MI455X (gfx1250) — compile-verified
//
#include <hip/hip_runtime.h>
#include <hip/hip_bf16.h>

#define HIDDEN 2048
#define NEXP   64
#define TOPK   8
#define KSPLIT 4
#define KCHUNK (HIDDEN / KSPLIT)   // 512

typedef float v2f __attribute__((ext_vector_type(2)));
typedef float v4f __attribute__((ext_vector_type(4)));
typedef float v8f __attribute__((ext_vector_type(8)));

// ---------------------------------------------------------------------------
// Kernel 1: logits = x @ W^T + b with V_WMMA_F32_16X16X4_F32.
// Block = 256 threads = 8 waves = 2 token-tiles x 4 K-chunks.
// Each wave computes a 16x64 partial over 512 columns with b128 loads:
//   lane half 0 loads x[row][k0..k0+3], half 1 loads x[row][k0+4..k0+7];
//   WMMA#1 consumes K-set {k0,k0+1,k0+4,k0+5} via (v.x,v.y),
//   WMMA#2 consumes K-set {k0+2,k0+3,k0+6,k0+7} via (v.z,v.w).
// Partials from K-chunks 1..3 are reduced through LDS (stride-33 layout,
// conflict-free across 32 lanes), chunk-0 waves add bias and store logits.
// ---------------------------------------------------------------------------
__global__ __launch_bounds__(256) void router_gemm_wmma(
    const float* __restrict__ x, const float* __restrict__ W,
    const float* __restrict__ bias, float* __restrict__ logits, int T)
{
    __shared__ float red[2 * (KSPLIT - 1) * 32 * 33];   // ~25 KB

    const int wave = threadIdx.x >> 5;            // 0..7
    const int lane = threadIdx.x & 31;
    const int tloc = wave & 1;                    // tile slot within block
    const int kc   = wave >> 1;                   // K-chunk 0..3
    int row0 = (blockIdx.x * 2 + tloc) * 16;
    if (row0 + 16 > T) row0 = T - 16;             // clamp (benign duplicate work)

    const int l16 = lane & 15;
    const int kh  = lane >> 4;                    // 0/1 -> K nibble {0..3} vs {4..7}
    const int kbase = kc * KCHUNK;

    const float* xp  = x + (size_t)(row0 + l16) * HIDDEN + kbase + 4 * kh;
    const float* wp0 = W + (size_t)(l16 +  0) * HIDDEN + kbase + 4 * kh;
    const float* wp1 = W + (size_t)(l16 + 16) * HIDDEN + kbase + 4 * kh;
    const float* wp2 = W + (size_t)(l16 + 32) * HIDDEN + kbase + 4 * kh;
    const float* wp3 = W + (size_t)(l16 + 48) * HIDDEN + kbase + 4 * kh;

    v8f acc0 = {}, acc1 = {}, acc2 = {}, acc3 = {};

    for (int kb = 0; kb < KCHUNK; kb += 64) {
        __builtin_prefetch(xp + kb + 256, 0, 3);  // stream-ahead, keep near
#pragma unroll
        for (int ku = 0; ku < 64; ku += 8) {
            const int k0 = kb + ku;
            v4f a  = *(const v4f*)(xp  + k0);
            v4f w0 = *(const v4f*)(wp0 + k0);
            v4f w1 = *(const v4f*)(wp1 + k0);
            v4f w2 = *(const v4f*)(wp2 + k0);
            v4f w3 = *(const v4f*)(wp3 + k0);
            v2f alo = __builtin_shufflevector(a, a, 0, 1);
            v2f ahi = __builtin_shufflevector(a, a, 2, 3);
            acc0 = __builtin_amdgcn_wmma_f32_16x16x4_f32(false, alo, false,
                       __builtin_shufflevector(w0, w0, 0, 1), (short)0, acc0, false, false);
            acc1 = __builtin_amdgcn_wmma_f32_16x16x4_f32(false, alo, false,
                       __builtin_shufflevector(w1, w1, 0, 1), (short)0, acc1, false, false);
            acc2 = __builtin_amdgcn_wmma_f32_16x16x4_f32(false, alo, false,
                       __builtin_shufflevector(w2, w2, 0, 1), (short)0, acc2, false, false);
            acc3 = __builtin_amdgcn_wmma_f32_16x16x4_f32(false, alo, false,
                       __builtin_shufflevector(w3, w3, 0, 1), (short)0, acc3, false, false);
            acc0 = __builtin_amdgcn_wmma_f32_16x16x4_f32(false, ahi, false,
                       __builtin_shufflevector(w0, w0, 2, 3), (short)0, acc0, false, false);
            acc1 = __builtin_amdgcn_wmma_f32_16x16x4_f32(false, ahi, false,
                       __builtin_shufflevector(w1, w1, 2, 3), (short)0, acc1, false, false);
            acc2 = __builtin_amdgcn_wmma_f32_16x16x4_f32(false, ahi, false,
                       __builtin_shufflevector(w2, w2, 2, 3), (short)0, acc2, false, false);
            acc3 = __builtin_amdgcn_wmma_f32_16x16x4_f32(false, ahi, false,
                       __builtin_shufflevector(w3, w3, 2, 3), (short)0, acc3, false, false);
        }
    }

    // -------- LDS reduction of the 4 K-chunk partials --------
    if (kc > 0) {
        float* dst = red + ((size_t)(tloc * (KSPLIT - 1) + (kc - 1)) * 32 + lane) * 33;
#pragma unroll
        for (int i = 0; i < 8; ++i) {
            dst[i]      = acc0[i];
            dst[8 + i]  = acc1[i];
            dst[16 + i] = acc2[i];
            dst[24 + i] = acc3[i];
        }
    }
    __syncthreads();
    if (kc != 0) return;

#pragma unroll
    for (int s = 0; s < KSPLIT - 1; ++s) {
        const float* src = red + ((size_t)(tloc * (KSPLIT - 1) + s) * 32 + lane) * 33;
#pragma unroll
        for (int i = 0; i < 8; ++i) {
            acc0[i] += src[i];
            acc1[i] += src[8 + i];
            acc2[i] += src[16 + i];
            acc3[i] += src[24 + i];
        }
    }

    // Epilogue: C/D layout -> VGPR r: lanes 0-15 hold M=r, lanes 16-31 M=8+r; N=l16.
    const float bv0 = bias[ 0 + l16];
    const float bv1 = bias[16 + l16];
    const float bv2 = bias[32 + l16];
    const float bv3 = bias[48 + l16];
    const int mrow = row0 + kh * 8;
#pragma unroll
    for (int r = 0; r < 8; ++r) {
        float* lrow = logits + (size_t)(mrow + r) * NEXP;
        lrow[ 0 + l16] = acc0[r] + bv0;
        lrow[16 + l16] = acc1[r] + bv1;
        lrow[32 + l16] = acc2[r] + bv2;
        lrow[48 + l16] = acc3[r] + bv3;
    }
}

// ---------------------------------------------------------------------------
// Kernel 2: per-token softmax + top-8 + renormalized gates (register-resident,
// fully unrolled; strict '>' reproduces jax.lax.top_k lowest-index tie-break).
// ---------------------------------------------------------------------------
__global__ __launch_bounds__(256) void router_topk_softmax(
    const float* __restrict__ logits, float* __restrict__ g_top,
    float* __restrict__ topk_out, int T)
{
    const int t = blockIdx.x * 256 + threadIdx.x;
    if (t >= T) return;

    float p[NEXP];
    const float4* lp = (const float4*)(logits + (size_t)t * NEXP);
#pragma unroll
    for (int i = 0; i < NEXP / 4; ++i) {
        float4 v = lp[i];
        p[4 * i + 0] = v.x; p[4 * i + 1] = v.y;
        p[4 * i + 2] = v.z; p[4 * i + 3] = v.w;
    }

    float mx = p[0];
#pragma unroll
    for (int e = 1; e < NEXP; ++e) mx = fmaxf(mx, p[e]);

    float S = 0.0f;
#pragma unroll
    for (int e = 0; e < NEXP; ++e) { p[e] = __expf(p[e] - mx); S += p[e]; }
    const float invS = 1.0f / S;          // g_soft[e] = p[e] * invS

    unsigned long long chosen = 0ull;
    float denom = 0.0f;                   // sum of top-k p[e]
    float idxf[TOPK];
#pragma unroll
    for (int k = 0; k < TOPK; ++k) {
        float best = -1.0f;
        int   bi   = 0;
#pragma unroll
        for (int e = 0; e < NEXP; ++e) {
            const bool avail = ((chosen >> e) & 1ull) == 0ull;
            const bool sel   = avail && (p[e] > best);
            best = sel ? p[e] : best;
            bi   = sel ? e    : bi;
        }
        chosen |= (1ull << bi);
        denom  += best;
        idxf[k] = (float)bi;
    }

    const float dg    = denom * invS + 1e-9f;
    const float scale = invS / dg;

    float4* gp = (float4*)(g_top + (size_t)t * NEXP);
#pragma unroll
    for (int i = 0; i < NEXP / 4; ++i) {
        float4 v;
        v.x = ((chosen >> (4 * i + 0)) & 1ull) ? p[4 * i + 0] * scale : 0.0f;
        v.y = ((chosen >> (4 * i + 1)) & 1ull) ? p[4 * i + 1] * scale : 0.0f;
        v.z = ((chosen >> (4 * i + 2)) & 1ull) ? p[4 * i + 2] * scale : 0.0f;
        v.w = ((chosen >> (4 * i + 3)) & 1ull) ? p[4 * i + 3] * scale : 0.0f;
        gp[i] = v;
    }

    float4* tp = (float4*)(topk_out + (size_t)t * TOPK);
    tp[0] = make_float4(idxf[0], idxf[1], idxf[2], idxf[3]);
    tp[1] = make_float4(idxf[4], idxf[5], idxf[6], idxf[7]);
}

// ---------------------------------------------------------------------------
extern "C" void kernel_launch(void* const* d_in, const int* in_sizes, int n_in,
                              void* d_out, int out_size, void* d_ws, size_t ws_size,
                              hipStream_t stream) {
    const float* x    = (const float*)d_in[0];   // [T, 2048]
    const float* W    = (const float*)d_in[1];   // [64, 2048]
    const float* bias = (const float*)d_in[2];   // [64]
    const int T = in_sizes[0] / HIDDEN;          // 16384

    float* out    = (float*)d_out;
    float* g_top  = out;                                  // [T, 64]
    float* topk   = out + (size_t)T * NEXP;               // [T, 8] (indices as floats)
    float* logits = topk + (size_t)T * TOPK;              // [T, 64]

    const int gemm_blocks = (T + 31) / 32;                // 2 tiles x 4 K-chunks per block
    router_gemm_wmma<<<gemm_blocks, 256, 0, stream>>>(x, W, bias, logits, T);

    router_topk_softmax<<<(T + 255) / 256, 256, 0, stream>>>(logits, g_top, topk, T);
}